// EventMessagePassingEdge_38740605010509
// MI455X (gfx1250) — compile-verified
//
#include <hip/hip_runtime.h>

// ---------------------------------------------------------------------------
// EventMessagePassingEdge for MI455X (gfx1250), wave32 + WMMA fp32 path.
//   evt = [h[src] | e_h | h[dst]]        (E x 192)
//   x   = evt @ W1 + b1                  (E x 64)
//   out = relu([x | ext] @ W2 + b2)      (E x 64)
// One wave computes a 16-edge x 64-out tile with V_WMMA_F32_16X16X4_F32.
// Weights are staged into LDS pre-swizzled into WMMA B-fragment order, so
// each B operand is a single conflict-free ds_load_b64 with an immediate
// offset (no address VALU, no regrouping movs).
// ---------------------------------------------------------------------------

#define IN_HID 64
#define EXT_DIM 32
#define K1 192              // 3*IN_HID
#define K2 96               // IN_HID + EXT_DIM
#define NOUT 64
#define KS1 (K1 / 4)        // 48 k-steps in GEMM1
#define KS2 (K2 / 4)        // 24 k-steps in GEMM2

#define LDX 100             // padded x-tile stride: conflict-free A frags / D spills
#define WAVES_PER_BLOCK 8
#define THREADS (WAVES_PER_BLOCK * 32)
#define EDGES_PER_BLOCK (WAVES_PER_BLOCK * 16)

typedef float v2f __attribute__((ext_vector_type(2)));
typedef float v4f __attribute__((ext_vector_type(4)));
typedef float v8f __attribute__((ext_vector_type(8)));

__global__ void __launch_bounds__(THREADS)
edge_mlp_wmma_kernel(const float* __restrict__ h,
                     const float* __restrict__ e_h,
                     const float* __restrict__ ext,
                     const float* __restrict__ W1,
                     const float* __restrict__ b1,
                     const float* __restrict__ W2,
                     const float* __restrict__ b2,
                     const int* __restrict__ src,
                     const int* __restrict__ dst,
                     float* __restrict__ out,
                     int E)
{
    // Weights in B-fragment order: frag f = (ks*4 + nt)*32 + lane holds the
    // float2 {W[k][n], W[k+1][n]} with k = 4*ks + 2*(lane>=16), n = nt*16 + (lane&15).
    __shared__ float sW1f[KS1 * 4 * 32 * 2];          // 48 KB
    __shared__ float sW2f[KS2 * 4 * 32 * 2];          // 24 KB
    __shared__ float sb1[NOUT];
    __shared__ float sb2[NOUT];
    __shared__ float sXT[WAVES_PER_BLOCK * 16 * LDX]; // 51.2 KB per-wave x|ext tiles

    const int tid = threadIdx.x;

    // ---- stage weights (fragment-swizzled) + biases into LDS ----
    for (int f = tid; f < KS1 * 4 * 32; f += THREADS) {
        const int l  = f & 31;
        const int nt = (f >> 5) & 3;
        const int ks = f >> 7;
        const int kk = (ks << 2) + ((l >> 4) << 1);
        const int n  = (nt << 4) + (l & 15);
        sW1f[2 * f]     = W1[kk * NOUT + n];
        sW1f[2 * f + 1] = W1[(kk + 1) * NOUT + n];
    }
    for (int f = tid; f < KS2 * 4 * 32; f += THREADS) {
        const int l  = f & 31;
        const int nt = (f >> 5) & 3;
        const int ks = f >> 7;
        const int kk = (ks << 2) + ((l >> 4) << 1);
        const int n  = (nt << 4) + (l & 15);
        sW2f[2 * f]     = W2[kk * NOUT + n];
        sW2f[2 * f + 1] = W2[(kk + 1) * NOUT + n];
    }
    if (tid < NOUT) { sb1[tid] = b1[tid]; sb2[tid] = b2[tid]; }
    __syncthreads();

    const int wave  = tid >> 5;
    const int lane  = tid & 31;
    const int ln    = lane & 15;          // M index (A/D fragments), N sub-index (B/D)
    const int khalf = (lane >> 4) << 1;   // 0 for lanes 0-15, 2 for lanes 16-31
    const int mhi   = (lane >> 4) << 3;   // 0 or 8: D-fragment row offset

    const int tileBase = blockIdx.x * EDGES_PER_BLOCK + wave * 16;
    if (tileBase >= E) return;            // whole-wave exit only (EXEC stays all-1s)

    // Per-lane row of the A fragment (clamped so tail lanes stay active).
    const int row  = tileBase + ln;
    const int rowc = row < E ? row : (E - 1);
    const int se = src[rowc];
    const int de = dst[rowc];
    const float* __restrict__ pS = h   + (size_t)se   * IN_HID;
    const float* __restrict__ pE = e_h + (size_t)rowc * IN_HID;
    const float* __restrict__ pD = h   + (size_t)de   * IN_HID;

    float* __restrict__ xt = sXT + wave * 16 * LDX;
    // Per-lane fragment bases: all subsequent B loads are base + immediate.
    const float* __restrict__ w1l = sW1f + lane * 2;
    const float* __restrict__ w2l = sW2f + lane * 2;

    // ---- stage ext columns (64..95) of the x2 tile early so the global
    //      loads overlap GEMM1. Lane l covers row (l&15), 16-col half (l>>4).
    {
        const int c0 = (lane >> 4) << 4;  // 0 or 16
        const float* __restrict__ pX = ext + (size_t)rowc * EXT_DIM + c0;
#pragma unroll
        for (int c = 0; c < 16; c += 4) {
            v4f v = *(const v4f*)(pX + c);
            *(v4f*)(xt + ln * LDX + IN_HID + c0 + c) = v;
        }
    }

    // ---- GEMM1: x[16x64] = evt[16x192] @ W1[192x64] ----
    v8f acc[4];
#pragma unroll
    for (int nt = 0; nt < 4; ++nt) acc[nt] = (v8f)0.0f;

    const float* __restrict__ segs[3] = { pS, pE, pD };
#pragma unroll
    for (int sgi = 0; sgi < 3; ++sgi) {
        const float* __restrict__ p = segs[sgi];
#pragma unroll
        for (int kk = 0; kk < IN_HID; kk += 4) {
            // A fragment 16x4 fp32: lane<16 -> K={kk,kk+1}; lane>=16 -> K={kk+2,kk+3}
            v2f a = *(const v2f*)(p + kk + khalf);
            const int ks = (sgi * IN_HID + kk) >> 2;
#pragma unroll
            for (int nt = 0; nt < 4; ++nt) {
                // One ds_load_b64 per fragment, immediate offset (ks*4+nt)*256 B.
                v2f b = *(const v2f*)(w1l + ((ks << 2) + nt) * 64);
                acc[nt] = __builtin_amdgcn_wmma_f32_16x16x4_f32(
                    false, a, false, b, (short)0, acc[nt], false, false);
            }
        }
    }

    // ---- epilogue 1: add b1, spill x-tile to LDS in row-major [M][k] ----
#pragma unroll
    for (int nt = 0; nt < 4; ++nt) {
        const float bb = sb1[nt * 16 + ln];
#pragma unroll
        for (int r = 0; r < 8; ++r) {
            // D layout: component r -> (M = r + mhi, N = nt*16 + ln)
            xt[(r + mhi) * LDX + nt * 16 + ln] = acc[nt][r] + bb;
        }
    }
    // Same-wave LDS write->read ordering is enforced via DScnt waits the
    // compiler inserts; xt is wave-private, so no barrier is needed.

    // ---- GEMM2: out[16x64] = x2[16x96] @ W2[96x64] ----
    v8f acc2[4];
#pragma unroll
    for (int nt = 0; nt < 4; ++nt) acc2[nt] = (v8f)0.0f;

#pragma unroll
    for (int kk = 0; kk < K2; kk += 4) {
        v2f a = *(const v2f*)(xt + ln * LDX + kk + khalf);
        const int ks = kk >> 2;
#pragma unroll
        for (int nt = 0; nt < 4; ++nt) {
            v2f b = *(const v2f*)(w2l + ((ks << 2) + nt) * 64);
            acc2[nt] = __builtin_amdgcn_wmma_f32_16x16x4_f32(
                false, a, false, b, (short)0, acc2[nt], false, false);
        }
    }

    // ---- epilogue 2: bias + ReLU + coalesced store ----
#pragma unroll
    for (int nt = 0; nt < 4; ++nt) {
        const float bb = sb2[nt * 16 + ln];
#pragma unroll
        for (int r = 0; r < 8; ++r) {
            int er = tileBase + r + mhi;
            if (er < E) {
                float v = acc2[nt][r] + bb;
                v = v > 0.0f ? v : 0.0f;
                out[(size_t)er * NOUT + nt * 16 + ln] = v;
            }
        }
    }
}

extern "C" void kernel_launch(void* const* d_in, const int* in_sizes, int n_in,
                              void* d_out, int out_size, void* d_ws, size_t ws_size,
                              hipStream_t stream) {
    const float* h   = (const float*)d_in[0];
    const float* e_h = (const float*)d_in[1];
    const float* ext = (const float*)d_in[2];
    const float* W1  = (const float*)d_in[3];
    const float* b1  = (const float*)d_in[4];
    const float* W2  = (const float*)d_in[5];
    const float* b2  = (const float*)d_in[6];
    const int*   src = (const int*)d_in[7];
    const int*   dst = (const int*)d_in[8];
    float* out = (float*)d_out;

    const int E = in_sizes[7];                       // number of edges
    const int blocks = (E + EDGES_PER_BLOCK - 1) / EDGES_PER_BLOCK;

    edge_mlp_wmma_kernel<<<blocks, THREADS, 0, stream>>>(
        h, e_h, ext, W1, b1, W2, b2, src, dst, out, E);
}